// GATEncoder_6803228196962
// MI455X (gfx1250) — compile-verified
//
#include <hip/hip_runtime.h>
#include <math.h>

typedef __attribute__((ext_vector_type(2))) float v2f;
typedef __attribute__((ext_vector_type(8))) float v8f;

#define NEG_SLOPE 0.2f

// Order-preserving monotone map fp32 -> uint32 so we can use atomicMax(u32)
// for the segment-max of attention logits (values can be negative).
__device__ __forceinline__ unsigned fenc(float f) {
  unsigned u = __float_as_uint(f);
  return (u & 0x80000000u) ? ~u : (u | 0x80000000u);
}
__device__ __forceinline__ float fdec(unsigned u) {
  return __uint_as_float((u & 0x80000000u) ? (u & 0x7fffffffu) : ~u);
}

__global__ void fill_u32_kernel(unsigned* __restrict__ p, unsigned v, int n) {
  int i = blockIdx.x * 256 + threadIdx.x;
  if (i < n) p[i] = v;
}
__global__ void fill_f32_kernel(float* __restrict__ p, float v, int n) {
  int i = blockIdx.x * 256 + threadIdx.x;
  if (i < n) p[i] = v;
}

// ---------------------------------------------------------------------------
// GEMM1: h1[N,128] = x[N,128] @ W1[128,128], fused alpha_src/alpha_dst (H=4,C=32)
// Block: 256 threads = 8 waves; block tile = 16 rows x 128 cols; wave tile 16x16.
// Exact fp32 via V_WMMA_F32_16X16X4_F32 (K=4 per op, 32 ops per wave-tile).
// LDS padding: Xs stride 132 -> A-frag reads hit 32 distinct banks;
//              Ws stride 136 -> B-frag lane-halves (rows k, k+2) 16 banks apart.
// ---------------------------------------------------------------------------
__global__ __launch_bounds__(256) void gemm1_kernel(
    const float* __restrict__ x, const float* __restrict__ W,
    const float* __restrict__ asrc, const float* __restrict__ adst,
    float* __restrict__ h, float* __restrict__ als, float* __restrict__ ald,
    int N)
{
  __shared__ float Ws[128 * 136];
  __shared__ float Xs[16 * 132];
  __shared__ float Cs[16 * 132];
  const int tid = threadIdx.x;

  for (int i = tid; i < 128 * 128; i += 256)
    Ws[(i >> 7) * 136 + (i & 127)] = W[i];

  const int row0 = blockIdx.x * 16;
  for (int i = tid; i < 16 * 128; i += 256) {
    int r = i >> 7, c = i & 127, gr = row0 + r;
    Xs[r * 132 + c] = (gr < N) ? x[(size_t)gr * 128 + c] : 0.0f;
  }
  __syncthreads();

  const int lane = tid & 31, wave = tid >> 5;
  const int nbase = wave * 16;           // wave's 16 output columns
  const int mn   = lane & 15;            // m (A rows) / n (B,C cols)
  const int kk   = (lane >> 4) << 1;     // lanes 0-15: k+0/1 ; lanes 16-31: k+2/3
  v8f acc = {};
  #pragma unroll
  for (int k0 = 0; k0 < 128; k0 += 4) {
    v2f a, b;
    a.x = Xs[mn * 132 + k0 + kk];
    a.y = Xs[mn * 132 + k0 + kk + 1];
    b.x = Ws[(k0 + kk)     * 136 + nbase + mn];
    b.y = Ws[(k0 + kk + 1) * 136 + nbase + mn];
    acc = __builtin_amdgcn_wmma_f32_16x16x4_f32(false, a, false, b,
                                                (short)0, acc, false, false);
  }
  // C layout: lane n = lane&15, rows m = j + 8*(lane>>4)
  const int half = lane >> 4;
  #pragma unroll
  for (int j = 0; j < 8; ++j)
    Cs[(j + 8 * half) * 132 + nbase + mn] = acc[j];
  __syncthreads();

  // coalesced tile store
  for (int i = tid; i < 16 * 128; i += 256) {
    int r = i >> 7, c = i & 127, gr = row0 + r;
    if (gr < N) h[(size_t)gr * 128 + c] = Cs[r * 132 + c];
  }
  // fused attention dots: alpha[n,h] = sum_c h[n,h*32+c] * a[h,c]
  if (tid < 64) {
    int r = tid >> 2, hh = tid & 3, gr = row0 + r;
    if (gr < N) {
      float ss = 0.f, sd = 0.f;
      #pragma unroll
      for (int c = 0; c < 32; ++c) {
        float hv = Cs[r * 132 + hh * 32 + c];
        ss = fmaf(hv, asrc[hh * 32 + c], ss);
        sd = fmaf(hv, adst[hh * 32 + c], sd);
      }
      als[gr * 4 + hh] = ss;
      ald[gr * 4 + hh] = sd;
    }
  }
}

// ---------------------------------------------------------------------------
// GEMM2: h2[N,64] = h1act[N,128] @ W2[128,64], fused alpha (H=1,C=64)
// Block tile = 32 rows x 64 cols (8 waves = 2 Mtiles x 4 Ntiles).
// ---------------------------------------------------------------------------
__global__ __launch_bounds__(256) void gemm2_kernel(
    const float* __restrict__ x, const float* __restrict__ W,
    const float* __restrict__ asrc, const float* __restrict__ adst,
    float* __restrict__ h, float* __restrict__ als, float* __restrict__ ald,
    int N)
{
  __shared__ float Ws[128 * 72];
  __shared__ float Xs[32 * 132];
  __shared__ float Cs[32 * 68];
  const int tid = threadIdx.x;

  for (int i = tid; i < 128 * 64; i += 256)
    Ws[(i >> 6) * 72 + (i & 63)] = W[i];

  const int row0 = blockIdx.x * 32;
  for (int i = tid; i < 32 * 128; i += 256) {
    int r = i >> 7, c = i & 127, gr = row0 + r;
    Xs[r * 132 + c] = (gr < N) ? x[(size_t)gr * 128 + c] : 0.0f;
  }
  __syncthreads();

  const int lane = tid & 31, wave = tid >> 5;
  const int mbase = (wave >> 2) * 16;
  const int nbase = (wave & 3) * 16;
  const int mn = lane & 15;
  const int kk = (lane >> 4) << 1;
  v8f acc = {};
  #pragma unroll
  for (int k0 = 0; k0 < 128; k0 += 4) {
    v2f a, b;
    a.x = Xs[(mbase + mn) * 132 + k0 + kk];
    a.y = Xs[(mbase + mn) * 132 + k0 + kk + 1];
    b.x = Ws[(k0 + kk)     * 72 + nbase + mn];
    b.y = Ws[(k0 + kk + 1) * 72 + nbase + mn];
    acc = __builtin_amdgcn_wmma_f32_16x16x4_f32(false, a, false, b,
                                                (short)0, acc, false, false);
  }
  const int half = lane >> 4;
  #pragma unroll
  for (int j = 0; j < 8; ++j)
    Cs[(mbase + j + 8 * half) * 68 + nbase + mn] = acc[j];
  __syncthreads();

  for (int i = tid; i < 32 * 64; i += 256) {
    int r = i >> 6, c = i & 63, gr = row0 + r;
    if (gr < N) h[(size_t)gr * 64 + c] = Cs[r * 68 + c];
  }
  if (tid < 32) {
    int gr = row0 + tid;
    if (gr < N) {
      float ss = 0.f, sd = 0.f;
      #pragma unroll
      for (int c = 0; c < 64; ++c) {
        float hv = Cs[tid * 68 + c];
        ss = fmaf(hv, asrc[c], ss);
        sd = fmaf(hv, adst[c], sd);
      }
      als[gr] = ss;
      ald[gr] = sd;
    }
  }
}

// ---------------------------------------------------------------------------
// Edge pass 1 (per layer): segment-max of leaky-relu logits via encoded u32 max.
// Self-loops handled implicitly: edges [E, E+N) are (i,i).
// ---------------------------------------------------------------------------
__global__ void edge_max1_kernel(const int* __restrict__ esrc, const int* __restrict__ edst,
                                 const float* __restrict__ als, const float* __restrict__ ald,
                                 unsigned* __restrict__ m, int E, int EN)
{
  int e = blockIdx.x * 256 + threadIdx.x;
  if (e >= EN) return;
  int s = (e < E) ? esrc[e] : (e - E);
  int d = (e < E) ? edst[e] : (e - E);
  #pragma unroll
  for (int h = 0; h < 4; ++h) {
    float v = als[s * 4 + h] + ald[d * 4 + h];
    v = (v > 0.f) ? v : NEG_SLOPE * v;
    atomicMax(&m[d * 4 + h], fenc(v));
  }
}

__global__ void edge_max2_kernel(const int* __restrict__ esrc, const int* __restrict__ edst,
                                 const float* __restrict__ als, const float* __restrict__ ald,
                                 unsigned* __restrict__ m, int E, int EN)
{
  int e = blockIdx.x * 256 + threadIdx.x;
  if (e >= EN) return;
  int s = (e < E) ? esrc[e] : (e - E);
  int d = (e < E) ? edst[e] : (e - E);
  float v = als[s] + ald[d];
  v = (v > 0.f) ? v : NEG_SLOPE * v;
  atomicMax(&m[d], fenc(v));
}

// ---------------------------------------------------------------------------
// Edge pass 2 (fused softmax-numerator + denominator + feature scatter).
// One wave per edge; lane j covers channels lane+32j -> head j (128B coalesced
// bursts for both the gather of h[src] and the L2-resident atomic scatter).
// Normalization by the denominator is deferred to the node-wise finalize.
// ---------------------------------------------------------------------------
__global__ __launch_bounds__(256) void edge_agg1_kernel(
    const int* __restrict__ esrc, const int* __restrict__ edst,
    const float* __restrict__ als, const float* __restrict__ ald,
    const unsigned* __restrict__ m, const float* __restrict__ hfeat,
    float* __restrict__ agg, float* __restrict__ denom, int E, int EN)
{
  int lane = threadIdx.x & 31;
  int e = blockIdx.x * 8 + (threadIdx.x >> 5);
  if (e >= EN) return;
  int s = (e < E) ? esrc[e] : (e - E);
  int d = (e < E) ? edst[e] : (e - E);
  float w[4];
  #pragma unroll
  for (int h = 0; h < 4; ++h) {
    float v = als[s * 4 + h] + ald[d * 4 + h];
    v = (v > 0.f) ? v : NEG_SLOPE * v;
    w[h] = __expf(v - fdec(m[d * 4 + h]));
  }
  if (lane < 4) atomicAdd(&denom[d * 4 + lane], w[lane]);
  #pragma unroll
  for (int j = 0; j < 4; ++j) {
    int ch = lane + 32 * j;   // head = ch/32 = j
    atomicAdd(&agg[(size_t)d * 128 + ch], hfeat[(size_t)s * 128 + ch] * w[j]);
  }
}

__global__ __launch_bounds__(256) void edge_agg2_kernel(
    const int* __restrict__ esrc, const int* __restrict__ edst,
    const float* __restrict__ als, const float* __restrict__ ald,
    const unsigned* __restrict__ m, const float* __restrict__ hfeat,
    float* __restrict__ agg, float* __restrict__ denom, int E, int EN)
{
  int lane = threadIdx.x & 31;
  int e = blockIdx.x * 8 + (threadIdx.x >> 5);
  if (e >= EN) return;
  int s = (e < E) ? esrc[e] : (e - E);
  int d = (e < E) ? edst[e] : (e - E);
  float v = als[s] + ald[d];
  v = (v > 0.f) ? v : NEG_SLOPE * v;
  float w = __expf(v - fdec(m[d]));
  if (lane == 0) atomicAdd(&denom[d], w);
  #pragma unroll
  for (int j = 0; j < 2; ++j) {
    int ch = lane + 32 * j;
    atomicAdd(&agg[(size_t)d * 64 + ch], hfeat[(size_t)s * 64 + ch] * w);
  }
}

// finalize layer1: normalize, +bias, ELU — in place on agg -> h1act
__global__ void finalize1_kernel(float* __restrict__ agg, const float* __restrict__ denom,
                                 const float* __restrict__ bias, int N)
{
  int i = blockIdx.x * 256 + threadIdx.x;
  if (i >= N * 128) return;
  int n = i >> 7, ch = i & 127;
  float v = agg[i] / (denom[n * 4 + (ch >> 5)] + 1e-16f) + bias[ch];
  agg[i] = (v > 0.f) ? v : (__expf(v) - 1.0f);
}

// finalize layer2: normalize, +bias — in place on d_out
__global__ void finalize2_kernel(float* __restrict__ out, const float* __restrict__ denom,
                                 const float* __restrict__ bias, int N)
{
  int i = blockIdx.x * 256 + threadIdx.x;
  if (i >= N * 64) return;
  int n = i >> 6, c = i & 63;
  out[i] = out[i] / (denom[n] + 1e-16f) + bias[c];
}

extern "C" void kernel_launch(void* const* d_in, const int* in_sizes, int n_in,
                              void* d_out, int out_size, void* d_ws, size_t ws_size,
                              hipStream_t stream)
{
  const float* x    = (const float*)d_in[0];
  const int*   ei   = (const int*)d_in[1];     // [2,E] edge_index
  const float* W1   = (const float*)d_in[2];
  const float* as1w = (const float*)d_in[3];
  const float* ad1w = (const float*)d_in[4];
  const float* b1   = (const float*)d_in[5];
  const float* W2   = (const float*)d_in[6];
  const float* as2w = (const float*)d_in[7];
  const float* ad2w = (const float*)d_in[8];
  const float* b2   = (const float*)d_in[9];
  float* out = (float*)d_out;

  const int N  = in_sizes[0] / 128;
  const int E  = in_sizes[1] / 2;
  const int EN = E + N;                    // edges + self-loops
  const int* esrc = ei;
  const int* edst = ei + E;

  // carve workspace
  char* ws = (char*)d_ws;
  size_t off = 0;
  auto alloc = [&](size_t bytes) -> void* {
    void* p = ws + off;
    off = (off + bytes + 255) & ~(size_t)255;
    return p;
  };
  float*    h1raw = (float*)alloc((size_t)N * 128 * 4);
  float*    agg1  = (float*)alloc((size_t)N * 128 * 4);  // becomes h1act in place
  float*    h2raw = (float*)alloc((size_t)N * 64 * 4);
  float*    als1  = (float*)alloc((size_t)N * 4 * 4);
  float*    ald1  = (float*)alloc((size_t)N * 4 * 4);
  unsigned* m1    = (unsigned*)alloc((size_t)N * 4 * 4);
  float*    den1  = (float*)alloc((size_t)N * 4 * 4);
  float*    als2  = (float*)alloc((size_t)N * 4);
  float*    ald2  = (float*)alloc((size_t)N * 4);
  unsigned* m2    = (unsigned*)alloc((size_t)N * 4);
  float*    den2  = (float*)alloc((size_t)N * 4);
  (void)ws_size; (void)n_in; (void)out_size;

  const unsigned NEGINF_ENC = 0x007fffffu;  // fenc(-inf)

  // deterministic in-graph init (d_out/d_ws are poisoned by the harness)
  fill_u32_kernel<<<(N * 4 + 255) / 256, 256, 0, stream>>>(m1, NEGINF_ENC, N * 4);
  fill_f32_kernel<<<(N * 4 + 255) / 256, 256, 0, stream>>>(den1, 0.f, N * 4);
  fill_f32_kernel<<<(N * 128 + 255) / 256, 256, 0, stream>>>(agg1, 0.f, N * 128);
  fill_u32_kernel<<<(N + 255) / 256, 256, 0, stream>>>(m2, NEGINF_ENC, N);
  fill_f32_kernel<<<(N + 255) / 256, 256, 0, stream>>>(den2, 0.f, N);
  fill_f32_kernel<<<(N * 64 + 255) / 256, 256, 0, stream>>>(out, 0.f, N * 64);

  // layer 1
  gemm1_kernel<<<(N + 15) / 16, 256, 0, stream>>>(x, W1, as1w, ad1w, h1raw, als1, ald1, N);
  edge_max1_kernel<<<(EN + 255) / 256, 256, 0, stream>>>(esrc, edst, als1, ald1, m1, E, EN);
  edge_agg1_kernel<<<(EN + 7) / 8, 256, 0, stream>>>(esrc, edst, als1, ald1, m1, h1raw, agg1, den1, E, EN);
  finalize1_kernel<<<(N * 128 + 255) / 256, 256, 0, stream>>>(agg1, den1, b1, N);

  // layer 2
  gemm2_kernel<<<(N + 31) / 32, 256, 0, stream>>>(agg1, W2, as2w, ad2w, h2raw, als2, ald2, N);
  edge_max2_kernel<<<(EN + 255) / 256, 256, 0, stream>>>(esrc, edst, als2, ald2, m2, E, EN);
  edge_agg2_kernel<<<(EN + 7) / 8, 256, 0, stream>>>(esrc, edst, als2, ald2, m2, h2raw, out, den2, E, EN);
  finalize2_kernel<<<(N * 64 + 255) / 256, 256, 0, stream>>>(out, den2, b2, N);
}